// LMH_90254442758290
// MI455X (gfx1250) — compile-verified
//
#include <hip/hip_runtime.h>

// Spiking-neuron temporal scan (T=16) over 4.19M independent fp32 chains.
// Memory-bound: 256 MiB in + 256 MiB out -> ~23 us floor at 23.3 TB/s HBM;
// ~0.6 GFLOP total compute is negligible.  No WMMA: there is no contraction
// dimension, matrix units cannot reduce the compulsory HBM traffic.
//
// gfx1250 data path used here:
//   - GLOBAL_LOAD_ASYNC_TO_LDS_B128 (ASYNCcnt) : each lane DMAs all 16
//     timestep inputs into a private LDS slot with zero VGPR landing cost
//     (16 x 512B = 8KB outstanding per wave, VGPR-independent prefetch depth).
//   - s_wait_asynccnt N : async loads complete in order, so waiting
//     asynccnt <= (15-t) guarantees timestep t is resident in LDS.
//   - ds_load_b128 to consume, NT global_store_b128 for the streamed-once
//     output.  fp32 state (v_d, v_s) lives in VGPRs for the whole chain.

typedef __attribute__((ext_vector_type(4))) float v4f;

#define SNN_T 16
#define BLK   256

__device__ __forceinline__ float snn_sigmoid(float z) {
    return 1.0f / (1.0f + __expf(-z));
}

__global__ __launch_bounds__(BLK)
void LMH_90254442758290_scan_async(const float* __restrict__ x,
                                   const float* __restrict__ p_a1,
                                   const float* __restrict__ p_b1,
                                   const float* __restrict__ p_a2,
                                   const float* __restrict__ p_b2,
                                   const float* __restrict__ p_vth,
                                   float* __restrict__ out,
                                   int n4)  // number of float4 chains (N/4)
{
    // 16 timestep slots x 256 lanes x 16B = 64 KB (of 320 KB/WGP).
    __shared__ v4f stage[SNN_T * BLK];

    const int tid = threadIdx.x;
    const int i   = blockIdx.x * BLK + tid;
    if (i >= n4) return;

    // Derived scalar parameters (uniform; scalar-float v_s_exp on gfx1250).
    const float a1  = snn_sigmoid(p_a1[0]) - 0.5f;
    const float b1  = snn_sigmoid(p_b1[0]) - 0.5f;
    const float a2  = snn_sigmoid(p_a2[0]) + 0.5f;
    const float b2  = snn_sigmoid(p_b2[0]) + 0.5f;
    const float vth = p_vth[0];

    const v4f* xp = (const v4f*)x + i;
    v4f*       op = (v4f*)out + i;

    // WG-relative LDS byte offset of this lane's t=0 slot (flat->LDS mapping
    // keeps the low 32 bits as the workgroup-relative offset).
    const unsigned lds0 = (unsigned)(size_t)(&stage[tid]);

    // ---- Issue phase: DMA all 16 timestep inputs into LDS (ASYNCcnt). ----
#pragma unroll
    for (int t = 0; t < SNN_T; ++t) {
        const v4f*     gp   = xp + (size_t)t * (size_t)n4;
        const unsigned loff = lds0 + (unsigned)(t * BLK * (int)sizeof(v4f));
        asm volatile("global_load_async_to_lds_b128 %0, %1, off"
                     :: "v"(loff), "v"(gp)
                     : "memory");
    }

    // ---- Consume phase: in-order completion => asynccnt <= (15-t). ----
    v4f vd = (v4f)(0.0f);
    v4f vs = (v4f)(0.5f * vth);

#pragma unroll
    for (int t = 0; t < SNN_T; ++t) {
        asm volatile("s_wait_asynccnt %0" :: "n"(SNN_T - 1 - t) : "memory");

        v4f xt = stage[t * BLK + tid];   // ds_load_b128, lane-private slot

        v4f o;
#pragma unroll
        for (int c = 0; c < 4; ++c) {
            float d  = fmaf(a1, vd[c], fmaf(b1, vs[c], xt[c]));  // v_d update
            float s  = fmaf(a2, vs[c], b2 * d);                  // v_s update
            float sp = (s >= vth) ? vth : 0.0f;                  // spike*vth
            vd[c] = d;
            vs[c] = s - sp;                                      // soft reset
            o[c]  = sp;
        }

        // Streamed-once output: 128-bit non-temporal store.
        __builtin_nontemporal_store(o, op + (size_t)t * (size_t)n4);
    }
}

// Scalar tail for N not divisible by 4 (not hit for the reference shape).
__global__ __launch_bounds__(BLK)
void LMH_90254442758290_scan_tail(const float* __restrict__ x,
                                  const float* __restrict__ p_a1,
                                  const float* __restrict__ p_b1,
                                  const float* __restrict__ p_a2,
                                  const float* __restrict__ p_b2,
                                  const float* __restrict__ p_vth,
                                  float* __restrict__ out,
                                  long base, long N, int rem)
{
    const int k = blockIdx.x * blockDim.x + threadIdx.x;
    if (k >= rem) return;
    const long i = base + (long)k;

    const float a1  = snn_sigmoid(p_a1[0]) - 0.5f;
    const float b1  = snn_sigmoid(p_b1[0]) - 0.5f;
    const float a2  = snn_sigmoid(p_a2[0]) + 0.5f;
    const float b2  = snn_sigmoid(p_b2[0]) + 0.5f;
    const float vth = p_vth[0];

    float vd = 0.0f;
    float vs = 0.5f * vth;

#pragma unroll
    for (int t = 0; t < SNN_T; ++t) {
        float xt = __builtin_nontemporal_load(x + (size_t)t * (size_t)N + i);
        float d  = fmaf(a1, vd, fmaf(b1, vs, xt));
        float s  = fmaf(a2, vs, b2 * d);
        float sp = (s >= vth) ? vth : 0.0f;
        vd = d;
        vs = s - sp;
        __builtin_nontemporal_store(sp, out + (size_t)t * (size_t)N + i);
    }
}

extern "C" void kernel_launch(void* const* d_in, const int* in_sizes, int n_in,
                              void* d_out, int out_size, void* d_ws, size_t ws_size,
                              hipStream_t stream) {
    const float* x     = (const float*)d_in[0];
    const float* p_a1  = (const float*)d_in[1];
    const float* p_b1  = (const float*)d_in[2];
    const float* p_a2  = (const float*)d_in[3];
    const float* p_b2  = (const float*)d_in[4];
    const float* p_vth = (const float*)d_in[5];
    float* out = (float*)d_out;

    const long total = (long)in_sizes[0];   // T * B * C * H * W
    const long N     = total / SNN_T;       // elements per timestep
    const long N4    = N / 4;               // float4 chains
    const long rem   = N - N4 * 4;

    if (N4 > 0) {
        dim3 block(BLK);
        dim3 grid((unsigned int)((N4 + BLK - 1) / BLK));
        LMH_90254442758290_scan_async<<<grid, block, 0, stream>>>(
            x, p_a1, p_b1, p_a2, p_b2, p_vth, out, (int)N4);
    }
    if (rem > 0) {
        dim3 block(BLK);
        dim3 grid((unsigned int)((rem + BLK - 1) / BLK));
        LMH_90254442758290_scan_tail<<<grid, block, 0, stream>>>(
            x, p_a1, p_b1, p_a2, p_b2, p_vth, out, N4 * 4, N, (int)rem);
    }
}